// PRNNLayer_52518860096610
// MI455X (gfx1250) — compile-verified
//
#include <hip/hip_runtime.h>
#include <hip/hip_bf16.h>

typedef float v2f __attribute__((ext_vector_type(2)));
typedef float v8f __attribute__((ext_vector_type(8)));

static constexpr int BB   = 2048;   // batch
static constexpr int TT   = 4096;   // time
static constexpr int TAPS = 30;

// ---------------------------------------------------------------------------
// tanh(x) for x >= 0 via one hardware exp:  tanh(x) = (1-e^{-2x})/(1+e^{-2x})
__device__ __forceinline__ float fast_tanh_pos(float x) {
  float e = __expf(-2.0f * x);
  return __fdividef(1.0f - e, 1.0f + e);
}

// x^{-1/4} for x > 0
__device__ __forceinline__ float quarter_inv_pow(float x) {
  return rsqrtf(sqrtf(x));
}

// ---------------------------------------------------------------------------
// Kernel A: compute PET and transpose (B,T,3) -> t-major {p, pet}[t*BB + b]
// 32x32 tile via LDS so both read and write sides are coalesced.
__global__ __launch_bounds__(256) void pet_transpose_kernel(
    const float* __restrict__ inp, float2* __restrict__ ppet) {
  __shared__ float2 tile[32][33];
  const int t0 = blockIdx.x * 32;
  const int b0 = blockIdx.y * 32;
  const int tx = threadIdx.x;   // 0..31
  const int ty = threadIdx.y;   // 0..7
#pragma unroll
  for (int i = 0; i < 4; ++i) {
    int b = b0 + ty + 8 * i;
    int t = t0 + tx;
    const float* e = inp + ((size_t)b * TT + t) * 3;
    float p    = e[0];
    float temp = e[1];
    float dayl = e[2];
    float pet = 29.8f * (dayl * 24.0f) * 0.611f *
                __expf(__fdividef(17.3f * temp, temp + 237.3f)) *
                __frcp_rn(temp + 273.2f);
    tile[tx][ty + 8 * i] = make_float2(p, pet);
  }
  __syncthreads();
#pragma unroll
  for (int i = 0; i < 4; ++i) {
    int t = t0 + ty + 8 * i;
    int b = b0 + tx;
    ppet[(size_t)t * BB + b] = tile[ty + 8 * i][tx];
  }
}

// ---------------------------------------------------------------------------
// Kernel B: GR4J sequential scan over T. One thread per batch row; all
// loads/stores are t-major so every wave access is a coalesced 128B line.
__global__ __launch_bounds__(32) void gr4j_scan_kernel(
    const float2* __restrict__ ppet,
    const float* __restrict__ x1p, const float* __restrict__ x2p,
    const float* __restrict__ x3p,
    float* __restrict__ PRt, float* __restrict__ Qtt) {
  const int b = blockIdx.x * 32 + threadIdx.x;

  const float X1 = fminf(fmaxf(x1p[0], 1.0f / 2000.0f), 1.0f) * 2000.0f;
  const float X2 = x2p[0] * 40.0f - 20.0f;
  const float X3 = fminf(fmaxf(x3p[0], 1.0f / 300.0f), 1.0f) * 300.0f;
  const float invX1 = __frcp_rn(X1);
  const float invX3 = __frcp_rn(X3);

  float s = 0.3f * X1;
  float r = 0.7f * X3;

  for (int t = 0; t < TT; ++t) {
    float2 pe = ppet[(size_t)t * BB + b];
    float P = pe.x, E = pe.y;
    bool  cond = P >= E;
    float Pn = cond ? (P - E) : 0.0f;
    float En = cond ? 0.0f : (E - P);

    float sn = s * invX1;
    float th_p = fast_tanh_pos(Pn * invX1);
    float Ps = cond ? X1 * (1.0f - sn * sn) * th_p *
                          __frcp_rn(1.0f + sn * th_p)
                    : 0.0f;
    float th_e = fast_tanh_pos(En * invX1);
    float Es = cond ? 0.0f
                    : s * (2.0f - sn) * th_e *
                          __frcp_rn(1.0f + (1.0f - sn) * th_e);
    s = fminf(fmaxf(s + Ps - Es, 0.0f), X1);

    float z  = (4.0f / 9.0f) * s * invX1;
    float z4 = (z * z) * (z * z);
    float Perc = s * (1.0f - quarter_inv_pow(1.0f + z4));
    s = fminf(fmaxf(s - Perc, 0.0f), X1);
    float PR = Perc + (Pn - Ps);

    float u  = r * invX3;                       // r >= 0 always
    float Fx = X2 * (u * u * u * sqrtf(u));     // u^3.5
    r = fmaxf(r + Fx, 0.0f);
    float w  = r * invX3;
    float w4 = (w * w) * (w * w);
    float Qr = r * (1.0f - quarter_inv_pow(1.0f + w4));
    r = fminf(fmaxf(r - Qr, 0.0f), X3);

    PRt[(size_t)t * BB + b] = PR;
    Qtt[(size_t)t * BB + b] = Qr;
  }
}

// ---------------------------------------------------------------------------
// Kernel C: fused 30-tap unit-hydrograph convolution as a banded matmul on
// the WMMA pipe + Qt add.  D(16x16) = A(16x48) x B(48x16) via 12x
// V_WMMA_F32_16X16X4_F32.  A = PR window (rows=batch), B = banded weights
// B[K][n] = comb[K-n].  D[i][n] = sum_k PR[b0+i][t0+n+k-14]*comb[k].
__global__ __launch_bounds__(256) void uh_conv_wmma_kernel(
    const float* __restrict__ PRt, const float* __restrict__ Qtt,
    const float* __restrict__ uh1, const float* __restrict__ uh2,
    float* __restrict__ out) {
  __shared__ float comb[32];
  const int tid = threadIdx.x;
  if (tid < 32) {
    float c = 0.0f;
    if (tid < TAPS) c = 0.9f * uh1[tid] + 0.1f * uh2[tid];
    comb[tid] = c;
  }
  __syncthreads();

  const int wave = tid >> 5;      // 0..7
  const int lane = tid & 31;
  const int half = lane >> 4;     // 0: K slots {0,1}; 1: K slots {2,3}
  const int l16  = lane & 15;

  const int NT = TT / 16;                       // 256 time tiles
  const int tileId = blockIdx.x * 8 + wave;     // 32768 tiles total
  const int b0 = (tileId / NT) * 16;
  const int t0 = (tileId % NT) * 16;

  // B fragments (banded weight matrix), per wmma layout:
  //   vgpr v, lanes 0-15: row K'=v (K'=v+2 for lanes 16-31), col n = lane&15
  v2f bfrag[12];
#pragma unroll
  for (int kk = 0; kk < 12; ++kk) {
#pragma unroll
    for (int v = 0; v < 2; ++v) {
      int K = 4 * kk + 2 * half + v;
      int k = K - l16;
      bfrag[kk][v] = (k >= 0 && k < TAPS) ? comb[k] : 0.0f;
    }
  }

  // Accumulate 12 K-slices.  Edge rows are handled with clamp+select so EXEC
  // stays all-ones around the WMMA ops (no divergent branches).
  v8f acc = {};
#pragma unroll
  for (int kk = 0; kk < 12; ++kk) {
    v2f a;
#pragma unroll
    for (int v = 0; v < 2; ++v) {
      int t   = t0 - 14 + (4 * kk + 2 * half + v);
      int tcl = min(max(t, 0), TT - 1);
      float val = PRt[(size_t)tcl * BB + b0 + l16];   // coalesced 64B/row
      a[v] = (t == tcl) ? val : 0.0f;
    }
    acc = __builtin_amdgcn_wmma_f32_16x16x4_f32(
        /*neg_a=*/false, a, /*neg_b=*/false, bfrag[kk],
        /*c_mod=*/(short)0, acc, /*reuse_a=*/false, /*reuse_b=*/false);
  }

  // Epilogue: D vgpr e, lanes 0-15 -> batch row e, lanes 16-31 -> row e+8;
  // col = time offset l16.  Stores are coalesced along t.
#pragma unroll
  for (int e = 0; e < 8; ++e) {
    int b = b0 + e + 8 * half;
    int t = t0 + l16;
    out[(size_t)b * TT + t] = acc[e] + Qtt[(size_t)t * BB + b];
  }
}

// ---------------------------------------------------------------------------
extern "C" void kernel_launch(void* const* d_in, const int* in_sizes, int n_in,
                              void* d_out, int out_size, void* d_ws,
                              size_t ws_size, hipStream_t stream) {
  const float* inp = (const float*)d_in[0];   // (B,T,3) f32
  const float* x1  = (const float*)d_in[1];
  const float* x2  = (const float*)d_in[2];
  const float* x3  = (const float*)d_in[3];
  const float* uh1 = (const float*)d_in[4];   // 30 f32
  const float* uh2 = (const float*)d_in[5];   // 30 f32
  float* out = (float*)d_out;                 // (B,T,1) f32

  const size_t N = (size_t)BB * TT;
  char* ws = (char*)d_ws;
  float2* ppet = (float2*)ws;                 // N * 8 bytes
  float*  PRt  = (float*)(ws + N * sizeof(float2));
  float*  Qtt  = PRt + N;                     // total: 16*N = 128 MiB

  pet_transpose_kernel<<<dim3(TT / 32, BB / 32), dim3(32, 8), 0, stream>>>(
      inp, ppet);
  gr4j_scan_kernel<<<BB / 32, 32, 0, stream>>>(ppet, x1, x2, x3, PRt, Qtt);
  uh_conv_wmma_kernel<<<(BB / 16) * (TT / 16) / 8, 256, 0, stream>>>(
      PRt, Qtt, uh1, uh2, out);
}